// DCTLayer_64338610094250
// MI455X (gfx1250) — compile-verified
//
#include <hip/hip_runtime.h>
#include <math.h>

typedef __attribute__((ext_vector_type(2))) float v2f;
typedef __attribute__((ext_vector_type(8))) float v8f;

static __device__ __forceinline__ v8f wmma_f32(v2f a, v2f b, v8f c) {
  // 8 args: (neg_a, A, neg_b, B, c_mod, C, reuse_a, reuse_b); f32 only allows C-modifiers.
  return __builtin_amdgcn_wmma_f32_16x16x4_f32(false, a, false, b, (short)0, c, false, false);
}

// cos(n*pi/4) exactly, branchless (values are only {0, +-sqrt(1/2), +-1}).
static __device__ __forceinline__ float cos8(int n) {
  n &= 7;
  const float R = 0.7071067811865476f;
  float mag = (n & 1) ? R : ((n & 2) ? 0.0f : 1.0f);
  float r = ((n >= 3) && (n <= 5)) ? -mag : mag;
  return r;
}
// sin(n*pi/4) = cos((n-2)*pi/4)
static __device__ __forceinline__ float sin8(int n) { return cos8(n + 6); }

// Zero the padded frequency planes f = 40..63 (reference zero-pads to N_FREQS=64).
__global__ void __launch_bounds__(256) dct_zero_tail(float* __restrict__ out) {
  const int per_b = 24 * 4096;              // 24 planes x 64x64 blocks
  int idx = blockIdx.x * 256 + threadIdx.x;
  if (idx < 32 * per_b) {
    int b = idx / per_b;
    int r = idx - b * per_b;
    out[b * (64 * 4096) + 40 * 4096 + r] = 0.0f;
  }
}

// One wave = one 16x16 spatial tile = 2x2 blocks of 8x8.
// Y = Fd * X * Fd^T with Fd = diag(F8,F8) = Cd + i*Sd, each scaled 1/sqrt(8) (ortho).
__global__ void __launch_bounds__(256) dct8x8_wmma(const float* __restrict__ x,
                                                   const float* __restrict__ fw,
                                                   float* __restrict__ out) {
  __shared__ float lds[8][2][256];          // per-wave T1,T2 bounce (16KB)
  const int lane = threadIdx.x & 31;
  const int wave = threadIdx.x >> 5;
  const int gw = blockIdx.x * 8 + wave;     // 32768 tiles exactly
  const int b  = gw >> 10;                  // batch 0..31
  const int ty = (gw >> 5) & 31;            // tile row 0..31
  const int tx = gw & 31;                   // tile col 0..31
  const int h  = lane >> 4;                 // lane half
  const int nl = lane & 15;                 // m (A-layout) / n (B,C/D-layout)

  // Block-diagonal DFT-8 constant chunks. C,S symmetric => A-layout chunk ==
  // B-layout chunk per lane: element [nl][4j + 2h + v].
  v2f ca[4], sa[4], nsa[4];
#pragma unroll
  for (int j = 0; j < 4; ++j) {
#pragma unroll
    for (int v = 0; v < 2; ++v) {
      const int k = 4 * j + 2 * h + v;
      float c = 0.0f, s = 0.0f;
      if ((nl >> 3) == (k >> 3)) {          // block-diagonal
        const int n = (nl & 7) * (k & 7);   // angle = n*pi/4
        c =  cos8(n) * 0.35355339059327373f;   // 1/sqrt(8)
        s = -sin8(n) * 0.35355339059327373f;   // F = cos - i*sin
      }
      ca[j][v] = c; sa[j][v] = s; nsa[j][v] = -s;
    }
  }

  // Load X tile as four B-operand chunks: chunk j row (4j+2h+v), col nl.
  const int r0 = ty * 16, c0 = tx * 16;
  const float* xb0 = x + (b * 512 + r0) * 512 + c0 + nl;
  v2f xb[4];
#pragma unroll
  for (int j = 0; j < 4; ++j) {
    const int rr = 4 * j + 2 * h;
    xb[j][0] = xb0[rr * 512];
    xb[j][1] = xb0[rr * 512 + 512];
  }

  // Pass 1: T1 = Cd*X, T2 = Sd*X   (8 wmmas)
  v8f t1 = {0, 0, 0, 0, 0, 0, 0, 0};
  v8f t2 = {0, 0, 0, 0, 0, 0, 0, 0};
#pragma unroll
  for (int j = 0; j < 4; ++j) {
    t1 = wmma_f32(ca[j], xb[j], t1);
    t2 = wmma_f32(sa[j], xb[j], t2);
  }

  // C/D layout -> LDS (row major 16x16) -> reload in A layout.
#pragma unroll
  for (int v = 0; v < 8; ++v) {
    lds[wave][0][(v + 8 * h) * 16 + nl] = t1[v];
    lds[wave][1][(v + 8 * h) * 16 + nl] = t2[v];
  }
  asm volatile("s_wait_dscnt 0" ::: "memory");  // LDS is in-order per wave

  // Pass 2: Y_re = T1*Cd^T - T2*Sd^T ; Y_im = T2*Cd^T + T1*Sd^T   (16 wmmas)
  v8f yre = {0, 0, 0, 0, 0, 0, 0, 0};
  v8f yim = {0, 0, 0, 0, 0, 0, 0, 0};
#pragma unroll
  for (int j = 0; j < 4; ++j) {
    const int off = nl * 16 + 4 * j + 2 * h;
    v2f a1; a1[0] = lds[wave][0][off]; a1[1] = lds[wave][0][off + 1];
    v2f a2; a2[0] = lds[wave][1][off]; a2[1] = lds[wave][1][off + 1];
    yre = wmma_f32(a1, ca[j],  yre);
    yre = wmma_f32(a2, nsa[j], yre);   // minus via pre-negated Sd
    yim = wmma_f32(a2, ca[j],  yim);
    yim = wmma_f32(a1, sa[j],  yim);
  }

  // C/D lane(h,nl), vgpr v holds block (ty*2+h, tx*2+nl/8), ky=v, kx=nl%8.
  const int kx = nl & 7;
  const int hb = ty * 2 + h;
  const int wb = tx * 2 + (nl >> 3);
  if (kx <= 4) {
    float* ob = out + b * (64 * 4096) + hb * 64 + wb;
#pragma unroll
    for (int v = 0; v < 8; ++v) {
      const int f = v * 5 + kx;                       // ky*5 + kx
      const float w = __fdividef(1.0f, 1.0f + __expf(-fw[f]));  // sigmoid
      const float mag = sqrtf(yre[v] * yre[v] + yim[v] * yim[v]);
      ob[f * 4096] = mag * w;
    }
  }
}

extern "C" void kernel_launch(void* const* d_in, const int* in_sizes, int n_in,
                              void* d_out, int out_size, void* d_ws, size_t ws_size,
                              hipStream_t stream) {
  (void)in_sizes; (void)n_in; (void)out_size; (void)d_ws; (void)ws_size;
  const float* x  = (const float*)d_in[0];   // (32,1,512,512) fp32
  const float* fw = (const float*)d_in[1];   // (64,) fp32
  float* out = (float*)d_out;                // (32,64,64,64) fp32

  dct_zero_tail<<<(32 * 24 * 4096 + 255) / 256, 256, 0, stream>>>(out);
  dct8x8_wmma<<<4096, 256, 0, stream>>>(x, fw, out);
}